// GIST_13761075216502
// MI455X (gfx1250) — compile-verified
//
#include <hip/hip_runtime.h>
#include <math.h>
#include <stdint.h>

typedef __attribute__((ext_vector_type(2))) float v2f;
typedef __attribute__((ext_vector_type(8))) float v8f;
typedef __attribute__((ext_vector_type(4))) unsigned int v4u;
typedef __attribute__((ext_vector_type(8))) int v8i;
typedef __attribute__((ext_vector_type(4))) int v4i;

#define S 320          // unified padded grid (266->320 zero-padded, and gist 320)
#define N1 266         // true prefilt FFT length
#define E (S*S)        // 102400 elements per image plane
#define IMGN 256
#define BATCH 16
#define PI_F 3.14159265358979323846f

#define BM 64
#define BN 64
#define KT 16
#define TILEF (BM*KT + KT*BN)   // floats per LDS stage buffer (2048)

#if defined(__has_builtin)
#if __has_builtin(__builtin_amdgcn_tensor_load_to_lds) && __has_builtin(__builtin_amdgcn_s_wait_tensorcnt)
#define USE_TDM 1
#endif
#endif
#ifndef USE_TDM
#define USE_TDM 0
#endif

// ---------------------------------------------------------------------------
// TDM: issue a 2-D tile load (tile_h rows x tile_w f32 elems, row pitch
// row_stride_elems) from global memory into LDS at byte offset lds_off.
// D# layout per CDNA5 ISA 8.3/8.4 (group0: count/lds/global/type,
// group1: data_size=4B, tensor dims, tile dims, dim0 stride).
// ---------------------------------------------------------------------------
#if USE_TDM
__device__ __forceinline__ void tdm_load_tile(unsigned lds_off, const void* gptr,
                                              unsigned tile_w, unsigned tile_h,
                                              unsigned row_stride_elems) {
    unsigned long long ga = (unsigned long long)(uintptr_t)gptr;
    v4u g0;
    g0.x = 1u;                                             // count=1, user desc
    g0.y = lds_off;                                        // LDS byte address
    g0.z = (unsigned)ga;                                   // global addr [31:0]
    g0.w = (unsigned)((ga >> 32) & 0x1FFFFFFull) | 0x80000000u; // [56:32] | type=2

    const unsigned td0 = 0x40000000u, td1 = 0x40000000u;   // huge tensor: no OOB clip
    unsigned long long s0 = (unsigned long long)row_stride_elems;
    v8i g1;
    g1[0] = (int)(2u << 16);                               // data_size=2 (4 bytes)
    g1[1] = (int)((td0 & 0xFFFFu) << 16);                  // tensor_dim0[15:0]
    g1[2] = (int)(((td0 >> 16) & 0xFFFFu) | ((td1 & 0xFFFFu) << 16));
    g1[3] = (int)(((td1 >> 16) & 0xFFFFu) | (tile_w << 16));   // tile_dim0
    g1[4] = (int)(tile_h & 0xFFFFu);                       // tile_dim1 | tile_dim2=0
    g1[5] = (int)(unsigned)(s0 & 0xFFFFFFFFull);           // dim0 stride [31:0]
    g1[6] = (int)(unsigned)((s0 >> 32) & 0xFFFFull);       // stride[47:32] | stride1 lo=0
    g1[7] = 0;
    v4i gz = {0, 0, 0, 0};
#if __clang_major__ >= 23
    v8i gz8 = {0, 0, 0, 0, 0, 0, 0, 0};
    __builtin_amdgcn_tensor_load_to_lds(g0, g1, gz, gz, gz8, 0);
#else
    __builtin_amdgcn_tensor_load_to_lds(g0, g1, gz, gz, 0);
#endif
}
#endif

// ---------------------------------------------------------------------------
// Batched real GEMM: C = alpha*A*B + beta*C.  M=N=K=S (320).
// 128 threads = 4 waves; block tile 64x64; each wave 32x32 (2x2 WMMA tiles).
// Double-buffered TDM pipeline: wave0 pre-issues the next K-tile's tensor
// loads into the alternate LDS buffer, then s_wait_tensorcnt(2) relies on
// in-order TENSORcnt completion to guarantee the current buffer landed.
// The dummy guarded store + asm fences keep the LDS reads live: the TDM
// writes LDS through an integer descriptor invisible to alias analysis.
// ---------------------------------------------------------------------------
__global__ __launch_bounds__(128)
void gemm_wmma_f32(float* __restrict__ C, const float* __restrict__ A,
                   const float* __restrict__ B,
                   long strideA, long strideB, long strideC,
                   float alpha, float beta) {
    __shared__ float smem[2 * TILEF];

    const int b = blockIdx.z;
    const float* Ab = A + (long)b * strideA;
    const float* Bb = B + (long)b * strideB;
    float*       Cb = C + (long)b * strideC;

    const int tn = blockIdx.x * BN;
    const int tm = blockIdx.y * BM;

    const int tid = threadIdx.x;
    const int wid = tid >> 5;
    const int l   = tid & 31;
    const int lm  = l & 15;
    const int hi  = l >> 4;
    const int wm  = (wid >> 1) * 32;
    const int wn  = (wid & 1) * 32;

    // Opaque, runtime-false store: gives smem a visible reaching store so the
    // compiler cannot fold the TDM-filled LDS reads to undef/zero.
    if (__builtin_expect(alpha == 12345.678f, 0)) smem[tid] = beta;

    v8f acc[2][2] = {};
    const int nIter = S / KT;

#if USE_TDM
    if (wid == 0) {   // preload K-tile 0 into buffer 0
        tdm_load_tile(0u, (const void*)(Ab + (long)tm * S), KT, BM, S);
        tdm_load_tile((unsigned)(BM * KT * 4), (const void*)(Bb + tn), BN, KT, S);
    }
#endif

    for (int it = 0; it < nIter; ++it) {
        const int cur = it & 1;
        const float* sA = smem + cur * TILEF;             // [64][16] row-major
        const float* sB = sA + BM * KT;                   // [16][64] row-major

#if USE_TDM
        asm volatile("" ::: "memory");   // pin prior reads before next TDM issue
        if (wid == 0) {
            if (it + 1 < nIter) {
                const int kn = (it + 1) * KT;
                const unsigned boff = (unsigned)((1 - cur) * TILEF * 4);
                tdm_load_tile(boff, (const void*)(Ab + (long)tm * S + kn), KT, BM, S);
                tdm_load_tile(boff + (unsigned)(BM * KT * 4),
                              (const void*)(Bb + (long)kn * S + tn), BN, KT, S);
                __builtin_amdgcn_s_wait_tensorcnt(2);   // oldest 2 (current tile) done
            } else {
                __builtin_amdgcn_s_wait_tensorcnt(0);
            }
        }
        __syncthreads();
        asm volatile("" ::: "memory");   // TDM wrote LDS: reads below must load
#else
        {
            const int k = it * KT;
            float* dA = smem + cur * TILEF;
            float* dB = dA + BM * KT;
            for (int i = tid; i < BM * KT; i += 128)
                dA[i] = Ab[(long)(tm + (i >> 4)) * S + (k + (i & 15))];
            for (int i = tid; i < KT * BN; i += 128)
                dB[i] = Bb[(long)(k + (i >> 6)) * S + (tn + (i & 63))];
            __syncthreads();
        }
#endif

#pragma unroll
        for (int kk = 0; kk < KT; kk += 4) {
            const int ka = kk + hi * 2;
            v2f a[2], bf[2];
#pragma unroll
            for (int mt = 0; mt < 2; ++mt) {
                const float* ap = &sA[(wm + mt * 16 + lm) * KT + ka];
                a[mt].x = ap[0]; a[mt].y = ap[1];
            }
#pragma unroll
            for (int nt = 0; nt < 2; ++nt) {
                const float* bp = &sB[ka * BN + wn + nt * 16 + lm];
                bf[nt].x = bp[0]; bf[nt].y = bp[BN];
            }
#pragma unroll
            for (int mt = 0; mt < 2; ++mt)
#pragma unroll
                for (int nt = 0; nt < 2; ++nt)
                    acc[mt][nt] = __builtin_amdgcn_wmma_f32_16x16x4_f32(
                        false, a[mt], false, bf[nt], (short)0, acc[mt][nt], false, false);
        }
        asm volatile("" ::: "memory");   // reads stay before the release barrier
        __syncthreads();
    }

    // C/D layout: VGPR r -> lanes 0-15: M=r, lanes 16-31: M=r+8; N = lane&15
#pragma unroll
    for (int mt = 0; mt < 2; ++mt) {
#pragma unroll
        for (int nt = 0; nt < 2; ++nt) {
            const int row0 = tm + wm + mt * 16 + hi * 8;
            const int col  = tn + wn + nt * 16 + lm;
#pragma unroll
            for (int r = 0; r < 8; ++r) {
                long idx = (long)(row0 + r) * S + col;
                float v = alpha * acc[mt][nt][r];
                if (beta != 0.0f) v += beta * Cb[idx];
                Cb[idx] = v;
            }
        }
    }
}

// ---------------------------------------------------------------------------
// DFT twiddle matrix, zero-padded to S x S:  W[k,n] = exp(-2*pi*i*k*n/N)
// ---------------------------------------------------------------------------
__global__ void make_dft(float* Wr, float* Wi, int N) {
    int idx = blockIdx.x * blockDim.x + threadIdx.x;
    if (idx >= E) return;
    int k = idx / S, n = idx % S;
    float wr = 0.0f, wi = 0.0f;
    if (k < N && n < N) {
        int m = (int)(((long)k * (long)n) % (long)N);
        float th = 6.283185307179586f * (float)m / (float)N;
        wr = cosf(th);
        wi = -sinf(th);
    }
    Wr[idx] = wr; Wi[idx] = wi;
}

// ---------------------------------------------------------------------------
// Channel mean + per-image min/max (one block per image)
// ---------------------------------------------------------------------------
__global__ void mean_minmax(const float* __restrict__ x, float* __restrict__ g,
                            float* __restrict__ mm) {
    const int b = blockIdx.x;
    const float* xb = x + (long)b * 3 * (IMGN * IMGN);
    float* gb = g + (long)b * (IMGN * IMGN);
    float mn = 3.0e38f, mx = -3.0e38f;
    for (int i = threadIdx.x; i < IMGN * IMGN; i += blockDim.x) {
        float v = (xb[i] + xb[i + IMGN * IMGN] + xb[i + 2 * IMGN * IMGN]) * (1.0f / 3.0f);
        gb[i] = v;
        mn = fminf(mn, v); mx = fmaxf(mx, v);
    }
    __shared__ float smn[256], smx[256];
    smn[threadIdx.x] = mn; smx[threadIdx.x] = mx;
    __syncthreads();
    for (int s = 128; s > 0; s >>= 1) {
        if (threadIdx.x < s) {
            smn[threadIdx.x] = fminf(smn[threadIdx.x], smn[threadIdx.x + s]);
            smx[threadIdx.x] = fmaxf(smx[threadIdx.x], smx[threadIdx.x + s]);
        }
        __syncthreads();
    }
    if (threadIdx.x == 0) { mm[2 * b] = smn[0]; mm[2 * b + 1] = smx[0]; }
}

// ---------------------------------------------------------------------------
// normalize -> log(1+x) -> symmetric pad w=5 into 266x266 region of 320x320
// ---------------------------------------------------------------------------
__global__ void prefilt_pad(const float* __restrict__ g, const float* __restrict__ mm,
                            float* __restrict__ P) {
    const int b = blockIdx.y;
    int idx = blockIdx.x * blockDim.x + threadIdx.x;
    if (idx >= E) return;
    int py = idx / S, px = idx % S;
    float v = 0.0f;
    if (py < N1 && px < N1) {
        int sy = (py < 5) ? (4 - py) : ((py < 261) ? (py - 5) : (516 - py));
        int sx = (px < 5) ? (4 - px) : ((px < 261) ? (px - 5) : (516 - px));
        float mn = mm[2 * b], mx = mm[2 * b + 1];
        float scale = 255.0f / (mx - mn);
        float gv = (g[(long)b * (IMGN * IMGN) + sy * IMGN + sx] - mn) * scale;
        v = logf(gv + 1.0f);
    }
    P[(long)b * E + idx] = v;
}

// ---------------------------------------------------------------------------
// Multiply spectrum by Gaussian transfer gf (computed on the fly), zero pads
// ---------------------------------------------------------------------------
__global__ void gf_mul(const float* __restrict__ Fr, const float* __restrict__ Fi,
                       float* __restrict__ Gr, float* __restrict__ Gi) {
    const int b = blockIdx.y;
    int idx = blockIdx.x * blockDim.x + threadIdx.x;
    if (idx >= E) return;
    int u = idx / S, v = idx % S;
    float gr = 0.0f, gi = 0.0f;
    if (u < N1 && v < N1) {
        float cu = (u < N1 / 2) ? (float)u : (float)(u - N1);
        float cv = (v < N1 / 2) ? (float)v : (float)(v - N1);
        // 1/s1^2 = ln2/16
        float gf = expf(-(cu * cu + cv * cv) * 0.04332169878499658f);
        long o = (long)b * E + idx;
        gr = Fr[o] * gf; gi = Fi[o] * gf;
    }
    Gr[(long)b * E + idx] = gr;
    Gi[(long)b * E + idx] = gi;
}

__global__ void sub_k(const float* __restrict__ A, const float* __restrict__ B,
                      float* __restrict__ C, long n) {
    long i = (long)blockIdx.x * blockDim.x + threadIdx.x;
    if (i < n) C[i] = A[i] - B[i];
}

__global__ void sq_k(const float* __restrict__ A, float* __restrict__ C, long n) {
    long i = (long)blockIdx.x * blockDim.x + threadIdx.x;
    if (i < n) { float v = A[i]; C[i] = v * v; }
}

// ---------------------------------------------------------------------------
// out = O/(0.2 + sqrt(|V|)) on the central 256x256, symmetric pad 32 -> 320x320
// ---------------------------------------------------------------------------
__global__ void finalize_pad(const float* __restrict__ O, const float* __restrict__ Vr,
                             const float* __restrict__ Vi, float* __restrict__ P2) {
    const int b = blockIdx.y;
    int idx = blockIdx.x * blockDim.x + threadIdx.x;
    if (idx >= E) return;
    int py = idx / S, px = idx % S;
    int sy = (py < 32) ? (31 - py) : ((py < 288) ? (py - 32) : (543 - py));
    int sx = (px < 32) ? (31 - px) : ((px < 288) ? (px - 32) : (543 - px));
    long q = (long)b * E + (long)(sy + 5) * S + (sx + 5);
    float vr = Vr[q], vi = Vi[q];
    float localstd = sqrtf(sqrtf(vr * vr + vi * vi));   // sqrt(abs(complex))
    P2[(long)b * E + idx] = O[q] / (0.2f + localstd);
}

// ---------------------------------------------------------------------------
// Multiply spectrum by Gabor transfer for filter f (computed on the fly)
// ---------------------------------------------------------------------------
__global__ void gab_mul(const float* __restrict__ Fr, const float* __restrict__ Fi,
                        float* __restrict__ Gr, float* __restrict__ Gi, int f) {
    int idx = blockIdx.x * blockDim.x + threadIdx.x;
    if (idx >= E) return;
    int u = idx / S, v = idx % S;
    float cu = (u < S / 2) ? (float)u : (float)(u - S);   // fy
    float cv = (v < S / 2) ? (float)v : (float)(v - S);   // fx
    float fr = sqrtf(cu * cu + cv * cv);
    float t  = atan2f(cu, cv);
    int i = f >> 3, j = f & 7;
    float p1 = 0.3f / powf(1.85f, (float)i);
    float tr = t + (PI_F / 8.0f) * (float)j;
    if (tr >  PI_F) tr -= 2.0f * PI_F;
    if (tr < -PI_F) tr += 2.0f * PI_F;
    float a = fr / ((float)S * p1) - 1.0f;
    float gab = expf(-3.5f * a * a - 2.0f * PI_F * tr * tr);
    for (int b = 0; b < BATCH; ++b) {
        long o = (long)b * E + idx;
        Gr[o] = Fr[o] * gab;
        Gi[o] = Fi[o] * gab;
    }
}

// ---------------------------------------------------------------------------
// |V| magnitude, 64x64 block means over the cropped [32:288) window
// out[b, f*16 + bx*4 + by]   (swapaxes(-1,-2) in reference)
// ---------------------------------------------------------------------------
__global__ void block_avg(const float* __restrict__ Vr, const float* __restrict__ Vi,
                          float* __restrict__ out, int f) {
    const int b   = blockIdx.x >> 4;
    const int blk = blockIdx.x & 15;
    const int by = blk >> 2, bx = blk & 3;
    const int y0 = 32 + by * 64, x0 = 32 + bx * 64;
    float s = 0.0f;
    for (int i = threadIdx.x; i < 4096; i += 256) {
        int yy = y0 + (i >> 6), xx = x0 + (i & 63);
        long o = (long)b * E + (long)yy * S + xx;
        float vr = Vr[o], vi = Vi[o];
        s += sqrtf(vr * vr + vi * vi);
    }
    __shared__ float sh[256];
    sh[threadIdx.x] = s;
    __syncthreads();
    for (int st = 128; st > 0; st >>= 1) {
        if (threadIdx.x < st) sh[threadIdx.x] += sh[threadIdx.x + st];
        __syncthreads();
    }
    if (threadIdx.x == 0)
        out[(long)b * 512 + f * 16 + bx * 4 + by] = sh[0] * (1.0f / 4096.0f);
}

// ---------------------------------------------------------------------------
// Host orchestration
// ---------------------------------------------------------------------------
extern "C" void kernel_launch(void* const* d_in, const int* in_sizes, int n_in,
                              void* d_out, int out_size, void* d_ws, size_t ws_size,
                              hipStream_t stream) {
    const float* x = (const float*)d_in[0];
    float* out = (float*)d_out;

    // ---- workspace layout (floats) ----
    float* p = (float*)d_ws;
    float* W1r = p; p += E;
    float* W1i = p; p += E;
    float* W2r = p; p += E;
    float* W2i = p; p += E;
    float* mm  = p; p += 32;
    float* g   = p; p += (long)BATCH * IMGN * IMGN;
    float* P   = p; p += (long)BATCH * E;
    float* A0  = p; p += (long)BATCH * E;
    float* A1  = p; p += (long)BATCH * E;
    float* A2  = p; p += (long)BATCH * E;
    float* A3  = p; p += (long)BATCH * E;
    float* A4  = p; p += (long)BATCH * E;
    float* A5  = p; p += (long)BATCH * E;

    const long sE = E;
    const float inv1 = 1.0f / ((float)N1 * (float)N1);
    const float inv2 = 1.0f / ((float)S * (float)S);

    auto G = [&](float* C, const float* A, const float* B,
                 long sA, long sB, float alpha, float beta) {
        dim3 grid(S / BN, S / BM, BATCH);
        gemm_wmma_f32<<<grid, 128, 0, stream>>>(C, A, B, sA, sB, sE, alpha, beta);
    };

    // twiddle matrices (W1 zero-padded 266->320)
    make_dft<<<(E + 255) / 256, 256, 0, stream>>>(W1r, W1i, N1);
    make_dft<<<(E + 255) / 256, 256, 0, stream>>>(W2r, W2i, S);

    // channel mean, min/max, normalize, log, pad -> P
    mean_minmax<<<BATCH, 256, 0, stream>>>(x, g, mm);
    prefilt_pad<<<dim3((E + 255) / 256, BATCH), 256, 0, stream>>>(g, mm, P);

    // ---- prefilt pass 1:  O = P - Re(ifft2(fft2(P)*gf)) ----
    G(A0, P, W1r, sE, 0, 1.0f, 0.0f);     // T = P*W1
    G(A1, P, W1i, sE, 0, 1.0f, 0.0f);
    G(A2, W1r, A0, 0, sE, 1.0f, 0.0f);    // F = W1*T
    G(A2, W1i, A1, 0, sE, -1.0f, 1.0f);
    G(A3, W1r, A1, 0, sE, 1.0f, 0.0f);
    G(A3, W1i, A0, 0, sE, 1.0f, 1.0f);
    gf_mul<<<dim3((E + 255) / 256, BATCH), 256, 0, stream>>>(A2, A3, A4, A5);
    G(A0, W1r, A4, 0, sE, 1.0f, 0.0f);    // U = conj(W1)*G
    G(A0, W1i, A5, 0, sE, 1.0f, 1.0f);
    G(A1, W1r, A5, 0, sE, 1.0f, 0.0f);
    G(A1, W1i, A4, 0, sE, -1.0f, 1.0f);
    G(A2, A0, W1r, sE, 0, inv1, 0.0f);    // Zr = Re(U*conj(W1))/N1^2
    G(A2, A1, W1i, sE, 0, inv1, 1.0f);
    {
        long n = (long)BATCH * E;
        sub_k<<<(int)((n + 255) / 256), 256, 0, stream>>>(P, A2, A3, n);  // O
        sq_k<<<(int)((n + 255) / 256), 256, 0, stream>>>(A3, A0, n);      // Q = O^2
    }

    // ---- prefilt pass 2: V = ifft2(fft2(O^2)*gf) ----
    G(A1, A0, W1r, sE, 0, 1.0f, 0.0f);
    G(A2, A0, W1i, sE, 0, 1.0f, 0.0f);
    G(A4, W1r, A1, 0, sE, 1.0f, 0.0f);
    G(A4, W1i, A2, 0, sE, -1.0f, 1.0f);
    G(A5, W1r, A2, 0, sE, 1.0f, 0.0f);
    G(A5, W1i, A1, 0, sE, 1.0f, 1.0f);
    gf_mul<<<dim3((E + 255) / 256, BATCH), 256, 0, stream>>>(A4, A5, A0, A1);
    G(A2, W1r, A0, 0, sE, 1.0f, 0.0f);
    G(A2, W1i, A1, 0, sE, 1.0f, 1.0f);
    G(A4, W1r, A1, 0, sE, 1.0f, 0.0f);
    G(A4, W1i, A0, 0, sE, -1.0f, 1.0f);
    G(A5, A2, W1r, sE, 0, inv1, 0.0f);    // Vr
    G(A5, A4, W1i, sE, 0, inv1, 1.0f);
    G(A0, A4, W1r, sE, 0, inv1, 0.0f);    // Vi
    G(A0, A2, W1i, sE, 0, -inv1, 1.0f);

    // normalize, crop [5:261), pad 32 -> 320x320  (A1 = img2)
    finalize_pad<<<dim3((E + 255) / 256, BATCH), 256, 0, stream>>>(A3, A5, A0, A1);

    // ---- gist: F2 = fft2(img2) ----
    G(A2, A1, W2r, sE, 0, 1.0f, 0.0f);
    G(A3, A1, W2i, sE, 0, 1.0f, 0.0f);
    G(A4, W2r, A2, 0, sE, 1.0f, 0.0f);    // F2r
    G(A4, W2i, A3, 0, sE, -1.0f, 1.0f);
    G(A5, W2r, A3, 0, sE, 1.0f, 0.0f);    // F2i
    G(A5, W2i, A2, 0, sE, 1.0f, 1.0f);

    // ---- per filter: V = ifft2(F2 .* gab_f); block means -> out ----
    for (int f = 0; f < 32; ++f) {
        gab_mul<<<(E + 255) / 256, 256, 0, stream>>>(A4, A5, A0, A1, f);
        G(A2, W2r, A0, 0, sE, 1.0f, 0.0f);   // U = conj(W2)*G
        G(A2, W2i, A1, 0, sE, 1.0f, 1.0f);
        G(A3, W2r, A1, 0, sE, 1.0f, 0.0f);
        G(A3, W2i, A0, 0, sE, -1.0f, 1.0f);
        G(A0, A2, W2r, sE, 0, inv2, 0.0f);   // Vr
        G(A0, A3, W2i, sE, 0, inv2, 1.0f);
        G(A1, A3, W2r, sE, 0, inv2, 0.0f);   // Vi
        G(A1, A2, W2i, sE, 0, -inv2, 1.0f);
        block_avg<<<BATCH * 16, 256, 0, stream>>>(A0, A1, out, f);
    }
    (void)in_sizes; (void)n_in; (void)out_size; (void)ws_size;
}